// GCN_10840497455133
// MI455X (gfx1250) — compile-verified
//
#include <hip/hip_runtime.h>
#include <hip/hip_bf16.h>

#define N_NODES 100000
#define N_EDGES 1600000
#define D_IN    128
#define D_HID   128
#define N_CLS   32
#define N_STRIPS (N_NODES / 16)   // 6250, exact

typedef float v2f __attribute__((ext_vector_type(2)));
typedef float v8f __attribute__((ext_vector_type(8)));

// ---------------------------------------------------------------------------
// Utility fills
// ---------------------------------------------------------------------------
__global__ void fill_f32(float* __restrict__ p, float v, long n) {
    long i = (long)blockIdx.x * 256 + threadIdx.x;
    if (i < n) p[i] = v;
}

// deg[dst] += 1 per edge (deg pre-filled with 1.0 for self-loops)
__global__ void deg_accum(const int* __restrict__ dst, float* __restrict__ deg) {
    int e = blockIdx.x * 256 + threadIdx.x;
    if (e < N_EDGES) unsafeAtomicAdd(&deg[dst[e]], 1.0f);
}

__global__ void rsqrt_kernel(const float* __restrict__ deg, float* __restrict__ dinv) {
    int i = blockIdx.x * 256 + threadIdx.x;
    if (i < N_NODES) {
        float d = deg[i];
        dinv[i] = d > 0.0f ? rsqrtf(d) : 0.0f;
    }
}

// ---------------------------------------------------------------------------
// WMMA GEMM with fused row-scale epilogue:
//   out[r, :] = (A[r, :] @ W) * dinv[r]
// A: [N_NODES, 128] row-major, W: [128, NCOLS] row-major.
// Block = 256 threads = 8 waves; each wave owns one 16-row strip.
// W staged in LDS in K-pair-interleaved form so every B fragment is one
// aligned ds_load_b64:  lW[(k>>1)*2N + 2n + (k&1)] = W[k][n].
// fp32 WMMA: V_WMMA_F32_16X16X4_F32, K-loop of 32.
// ---------------------------------------------------------------------------
template <int NCOLS>
__global__ __launch_bounds__(256)
void gemm_scale_kernel(const float* __restrict__ A,
                       const float* __restrict__ W,
                       const float* __restrict__ dinv,
                       float* __restrict__ out) {
    constexpr int NT = NCOLS / 16;
    __shared__ float lW[128 * NCOLS];

    const int tid = threadIdx.x;
    // cooperative stage of W into LDS, K-pair interleaved
    for (int idx = tid; idx < 128 * NCOLS; idx += 256) {
        const int k = idx / NCOLS;
        const int n = idx - k * NCOLS;
        lW[(k >> 1) * (2 * NCOLS) + 2 * n + (k & 1)] = W[idx];
    }
    __syncthreads();

    const int wave  = tid >> 5;
    const int lane  = tid & 31;
    const int strip = blockIdx.x * 8 + wave;
    if (strip >= N_STRIPS) return;

    const int row0 = strip * 16;
    const int half = lane >> 4;     // 0: lanes 0-15, 1: lanes 16-31
    const int l16  = lane & 15;
    const int arow = row0 + l16;    // A-matrix row for this lane

    v8f acc[NT];
#pragma unroll
    for (int t = 0; t < NT; ++t) acc[t] = (v8f){0.f, 0.f, 0.f, 0.f, 0.f, 0.f, 0.f, 0.f};

    // K loop: 4 K-values per WMMA.
    // A frag (16x4 f32 layout): VGPR0 = K(k0+2h), VGPR1 = K(k0+2h+1)
    for (int k0 = 0; k0 < 128; k0 += 4) {
        const int ka = k0 + 2 * half;          // always even
        const v2f a = *(const v2f*)&A[(long)arow * 128 + ka];
        const float* __restrict__ lrow = &lW[(ka >> 1) * (2 * NCOLS)];
#pragma unroll
        for (int t = 0; t < NT; ++t) {
            const int n = t * 16 + l16;
            const v2f b = *(const v2f*)&lrow[2 * n];   // (W[ka][n], W[ka+1][n])
            acc[t] = __builtin_amdgcn_wmma_f32_16x16x4_f32(
                false, a, false, b, (short)0, acc[t], false, false);
        }
    }

    // Epilogue: C/D layout -> VGPR r holds (M = r + 8*half, N = l16).
    // Fuse the dinv[src] half of the symmetric normalization here.
#pragma unroll
    for (int t = 0; t < NT; ++t) {
#pragma unroll
        for (int r = 0; r < 8; ++r) {
            const int rr = row0 + r + 8 * half;
            out[(long)rr * NCOLS + t * 16 + l16] = acc[t][r] * dinv[rr];
        }
    }
}

// ---------------------------------------------------------------------------
// Edge aggregation: t[dst, :] += hs[src, :]  (hs already scaled by dinv[src])
// One wave per edge.
// ---------------------------------------------------------------------------
__global__ __launch_bounds__(256)
void scatter128(const int* __restrict__ src, const int* __restrict__ dst,
                const float* __restrict__ hs, float* __restrict__ t1) {
    const int e = blockIdx.x * 8 + (threadIdx.x >> 5);
    const int lane = threadIdx.x & 31;
    if (e >= N_EDGES) return;
    const long s = src[e];
    const long d = dst[e];
    const float* __restrict__ hrow = hs + s * 128;
    float* __restrict__ trow = t1 + d * 128;
#pragma unroll
    for (int c = 0; c < 4; ++c) {
        unsafeAtomicAdd(&trow[lane + 32 * c], hrow[lane + 32 * c]);
    }
}

__global__ __launch_bounds__(256)
void scatter32(const int* __restrict__ src, const int* __restrict__ dst,
               const float* __restrict__ hs, float* __restrict__ t2) {
    const int e = blockIdx.x * 8 + (threadIdx.x >> 5);
    const int lane = threadIdx.x & 31;
    if (e >= N_EDGES) return;
    const long s = src[e];
    const long d = dst[e];
    unsafeAtomicAdd(&t2[d * 32 + lane], hs[s * 32 + lane]);
}

// ---------------------------------------------------------------------------
// Layer-1 post-pass: out = relu(dinv[n] * (t1 + hs) + b1), in place into t1.
// (+ hs implements the self-loop; * dinv[n] is the dst half of the norm.)
// ---------------------------------------------------------------------------
__global__ void post1(const float* __restrict__ hs, float* __restrict__ t1,
                      const float* __restrict__ dinv, const float* __restrict__ b1) {
    long i = (long)blockIdx.x * 256 + threadIdx.x;
    if (i >= (long)N_NODES * 128) return;
    const int node = (int)(i >> 7);
    const int c = (int)(i & 127);
    float v = (t1[i] + hs[i]) * dinv[node] + b1[c];
    t1[i] = fmaxf(v, 0.0f);
}

// ---------------------------------------------------------------------------
// Final: v = dinv[n]*(t2 + h2s) + b2; out = log_softmax(v) over 32 classes.
// One wave per node; wave32 shuffle reductions.
// ---------------------------------------------------------------------------
__global__ __launch_bounds__(256)
void final_kernel(const float* __restrict__ h2s, const float* __restrict__ t2,
                  const float* __restrict__ dinv, const float* __restrict__ b2,
                  float* __restrict__ out) {
    const int node = blockIdx.x * 8 + (threadIdx.x >> 5);
    const int lane = threadIdx.x & 31;
    if (node >= N_NODES) return;
    const long idx = (long)node * 32 + lane;
    float v = (t2[idx] + h2s[idx]) * dinv[node] + b2[lane];

    float m = v;
#pragma unroll
    for (int o = 16; o > 0; o >>= 1) m = fmaxf(m, __shfl_xor(m, o, 32));
    float s = __expf(v - m);
#pragma unroll
    for (int o = 16; o > 0; o >>= 1) s += __shfl_xor(s, o, 32);
    out[idx] = (v - m) - __logf(s);
}

// ---------------------------------------------------------------------------
// Orchestration
// ---------------------------------------------------------------------------
extern "C" void kernel_launch(void* const* d_in, const int* in_sizes, int n_in,
                              void* d_out, int out_size, void* d_ws, size_t ws_size,
                              hipStream_t stream) {
    const float* x  = (const float*)d_in[0];   // [N_NODES, 128]
    const int*   ei = (const int*)d_in[1];     // [2, N_EDGES]
    const float* W1 = (const float*)d_in[2];   // [128, 128]
    const float* b1 = (const float*)d_in[3];   // [128]
    const float* W2 = (const float*)d_in[4];   // [128, 32]
    const float* b2 = (const float*)d_in[5];   // [32]
    float* out = (float*)d_out;                // [N_NODES, 32]

    const int* src = ei;
    const int* dst = ei + N_EDGES;

    // Workspace layout (floats): deg | dinv | hs | t1 | h2s | t2
    float* deg  = (float*)d_ws;
    float* dinv = deg  + N_NODES;
    float* hs   = dinv + N_NODES;                    // [N_NODES, 128]
    float* t1   = hs   + (long)N_NODES * 128;        // [N_NODES, 128]
    float* h2s  = t1   + (long)N_NODES * 128;        // [N_NODES, 32]
    float* t2   = h2s  + (long)N_NODES * 32;         // [N_NODES, 32]

    const long n128 = (long)N_NODES * 128;
    const long n32  = (long)N_NODES * 32;

    // Degrees (self-loop => init 1.0), then dinv = rsqrt(deg)
    fill_f32<<<(N_NODES + 255) / 256, 256, 0, stream>>>(deg, 1.0f, N_NODES);
    deg_accum<<<(N_EDGES + 255) / 256, 256, 0, stream>>>(dst, deg);
    rsqrt_kernel<<<(N_NODES + 255) / 256, 256, 0, stream>>>(deg, dinv);

    // Zero accumulation buffers (must re-zero every call)
    fill_f32<<<(int)((n128 + 255) / 256), 256, 0, stream>>>(t1, 0.0f, n128);
    fill_f32<<<(int)((n32 + 255) / 256), 256, 0, stream>>>(t2, 0.0f, n32);

    // Layer 1: hs = (x @ W1) * dinv[row]   (WMMA fp32)
    gemm_scale_kernel<128><<<(N_STRIPS + 7) / 8, 256, 0, stream>>>(x, W1, dinv, hs);
    // t1[dst] += hs[src] over edges
    scatter128<<<(N_EDGES + 7) / 8, 256, 0, stream>>>(src, dst, hs, t1);
    // t1 = relu(dinv * (t1 + hs) + b1)   (in place; becomes layer-2 input)
    post1<<<(int)((n128 + 255) / 256), 256, 0, stream>>>(hs, t1, dinv, b1);

    // Layer 2: h2s = (t1 @ W2) * dinv[row]   (WMMA fp32)
    gemm_scale_kernel<32><<<(N_STRIPS + 7) / 8, 256, 0, stream>>>(t1, W2, dinv, h2s);
    // t2[dst] += h2s[src] over edges
    scatter32<<<(N_EDGES + 7) / 8, 256, 0, stream>>>(src, dst, h2s, t2);
    // out = log_softmax(dinv * (t2 + h2s) + b2)
    final_kernel<<<(N_NODES + 7) / 8, 256, 0, stream>>>(h2s, t2, dinv, b2, out);
}